// HeterogeneousGraphDiffusion_23287312679289
// MI455X (gfx1250) — compile-verified
//
#include <hip/hip_runtime.h>
#include <math.h>

// Problem constants (match reference)
#define N_NODES 1024
#define F_DIMS  512
#define T_STEPS 128
#define E_EDGES 32768
#define H_DIMS  512

typedef __attribute__((ext_vector_type(2))) float v2f;
typedef __attribute__((ext_vector_type(8))) float v8f;

// ---------------------------------------------------------------------------
// CDNA5 async global->LDS copy (ASYNCcnt-tracked, no VGPR round trip).
// LDS operand is the 32-bit LDS byte offset (low 32 bits of a generic
// pointer to shared memory, per ISA 10.2 aperture mapping).
// ---------------------------------------------------------------------------
__device__ __forceinline__ void async_load_b128(const float* gptr, const void* lptr) {
  unsigned int lds_off = (unsigned int)(uintptr_t)lptr;
  asm volatile("global_load_async_to_lds_b128 %0, %1, off"
               :: "v"(lds_off), "v"(gptr) : "memory");
}
__device__ __forceinline__ void wait_async0() {
  asm volatile("s_wait_asynccnt 0x0" ::: "memory");
}

// ---------------------------------------------------------------------------
// Diffusion schedules: sigmoid betas, cumprod alphas_bar (fp64 like numpy)
// ---------------------------------------------------------------------------
__global__ void sched_kernel(float* __restrict__ sab, float* __restrict__ somab) {
  __shared__ double betas[T_STEPS];
  const int i = threadIdx.x;
  if (i < T_STEPS) {
    double z = -6.0 + 12.0 * (double)i / (double)(T_STEPS - 1);
    double sig = 1.0 / (1.0 + exp(-z));
    betas[i] = sig * (0.02 - 1e-4) + 1e-4;
  }
  __syncthreads();
  if (i == 0) {
    double ab = 1.0;
    for (int t = 0; t < T_STEPS; ++t) {
      ab *= (1.0 - betas[t]);
      sab[t] = (float)sqrt(ab);
      somab[t] = (float)sqrt(1.0 - ab);
    }
  }
}

// ---------------------------------------------------------------------------
// LayerNorm of x and noise; nz = sign(xn) * |sqrt(2)*LN(noise)|.
// Seeds sel rows for t[i]==0 (diffused[0] == xn).
// ---------------------------------------------------------------------------
__global__ __launch_bounds__(256)
void ln_kernel(const float* __restrict__ x, const float* __restrict__ noise,
               const int* __restrict__ t, float* __restrict__ xn,
               float* __restrict__ nz, float* __restrict__ sel) {
  const int row = blockIdx.x;
  const int tid = threadIdx.x;
  const float* xr = x + (size_t)row * F_DIMS;
  const float* nr = noise + (size_t)row * F_DIMS;

  const float x0 = xr[tid], x1 = xr[tid + 256];
  const float n0 = nr[tid], n1 = nr[tid + 256];

  __shared__ float rsx[256], rsx2[256], rsn[256], rsn2[256];
  rsx[tid] = x0 + x1;           rsx2[tid] = x0 * x0 + x1 * x1;
  rsn[tid] = n0 + n1;           rsn2[tid] = n0 * n0 + n1 * n1;
  __syncthreads();
  for (int s = 128; s > 0; s >>= 1) {
    if (tid < s) {
      rsx[tid] += rsx[tid + s];   rsx2[tid] += rsx2[tid + s];
      rsn[tid] += rsn[tid + s];   rsn2[tid] += rsn2[tid + s];
    }
    __syncthreads();
  }
  __shared__ float bx[2], bn[2];
  if (tid == 0) {
    float mux = rsx[0] * (1.0f / F_DIMS);
    float varx = rsx2[0] * (1.0f / F_DIMS) - mux * mux;
    bx[0] = mux;  bx[1] = 1.0f / sqrtf(varx + 1e-5f);
    float mun = rsn[0] * (1.0f / F_DIMS);
    float varn = rsn2[0] * (1.0f / F_DIMS) - mun * mun;
    bn[0] = mun;  bn[1] = 1.0f / sqrtf(varn + 1e-5f);
  }
  __syncthreads();
  const float mux = bx[0], ivx = bx[1], mun = bn[0], ivn = bn[1];
  const int t0 = (t[row] == 0);
  const float SQ2 = 1.41421356237309515f;
#pragma unroll
  for (int j = 0; j < 2; ++j) {
    const int c = tid + j * 256;
    const float xv = ((j == 0 ? x0 : x1) - mux) * ivx;
    const float nv = ((j == 0 ? n0 : n1) - mun) * ivn;
    const float sgn = (xv > 0.0f) ? 1.0f : ((xv < 0.0f) ? -1.0f : 0.0f);
    const float nzv = sgn * fabsf(SQ2 * nv);
    const size_t idx = (size_t)row * F_DIMS + c;
    xn[idx] = xv;
    nz[idx] = nzv;
    if (t0) sel[idx] = xv;
  }
}

// ---------------------------------------------------------------------------
// Adjacency / Laplacian helpers
// ---------------------------------------------------------------------------
__global__ void zero_kernel(float* __restrict__ p, int n) {
  int i = blockIdx.x * blockDim.x + threadIdx.x;
  if (i < n) p[i] = 0.0f;
}

__global__ void eye_kernel(float* __restrict__ p) {
  int i = blockIdx.x * blockDim.x + threadIdx.x;
  int r = i >> 10;       // /1024
  int c = i & 1023;
  p[i] = (r == c) ? 1.0f : 0.0f;
}

__global__ void scatter_kernel(const int* __restrict__ src,
                               const int* __restrict__ dst,
                               float* __restrict__ adj) {
  int e = blockIdx.x * blockDim.x + threadIdx.x;
  if (e < E_EDGES) adj[(size_t)src[e] * N_NODES + dst[e]] = 1.0f;
}

// In-place: A <- adj/n - diag(rowsum(adj)/n)   (i.e. negL). Block per row.
__global__ __launch_bounds__(256)
void laplacian_kernel(float* __restrict__ A) {
  const int row = blockIdx.x;
  const int tid = threadIdx.x;
  float v[4];
  float s = 0.0f;
#pragma unroll
  for (int j = 0; j < 4; ++j) {
    v[j] = A[(size_t)row * N_NODES + tid + j * 256];
    s += v[j];
  }
  __shared__ float red[256];
  red[tid] = s;
  __syncthreads();
  for (int st = 128; st > 0; st >>= 1) {
    if (tid < st) red[tid] += red[tid + st];
    __syncthreads();
  }
  const float rowsum = red[0];
  const float inv_n = 1.0f / (float)N_NODES;
#pragma unroll
  for (int j = 0; j < 4; ++j) {
    const int col = tid + j * 256;
    float out = v[j] * inv_n;
    if (col == row) out -= rowsum * inv_n;
    A[(size_t)row * N_NODES + col] = out;
  }
}

// ---------------------------------------------------------------------------
// FP32 WMMA GEMM: C[M x Nc] = A[M x K] * B[K x Nc], row-major.
// Block: 256 threads = 8 waves, block tile 128x64, wave tile 32x32
// (2x2 WMMA 16x16 tiles). K staged in chunks of 32 through DOUBLE-BUFFERED
// LDS using GLOBAL_LOAD_ASYNC_TO_LDS_B128 (ASYNCcnt): the next chunk's
// async copy is issued right after the barrier and overlaps the current
// chunk's V_WMMA_F32_16X16X4_F32 compute. One barrier per iteration: it
// both publishes the arrived buffer and proves the other buffer's readers
// are done before it is overwritten.
//
// LDS strides: lsA 36 floats (144B, 16B-aligned; conflict-free column
// reads), lsB 80 floats (320B; half-wave row pairs 32 banks apart).
// ---------------------------------------------------------------------------
#define BLK_M 128
#define BLK_N 64
#define BLK_K 32
#define LDA_S (BLK_K + 4)    // 36 floats
#define LDB_S (BLK_N + 16)   // 80 floats

__global__ __launch_bounds__(256)
void wmma_mm_kernel(float* __restrict__ C, const float* __restrict__ A,
                    const float* __restrict__ B, int M, int Nc, int K,
                    const int* __restrict__ tsel, int step,
                    float* __restrict__ sel) {
  __shared__ float lsA[2][BLK_M][LDA_S];
  __shared__ float lsB[2][BLK_K][LDB_S];

  const int tid = threadIdx.x;
  const int lane = tid & 31;
  const int wave = tid >> 5;       // 0..7
  const int waveRow = wave >> 1;   // 0..3 -> 32-row strip
  const int waveCol = wave & 1;    // 0..1 -> 32-col strip
  const int l16 = lane & 15;
  const int half = lane >> 4;      // selects K pair within a WMMA step

  const int blockM = blockIdx.y * BLK_M;
  const int blockN = blockIdx.x * BLK_N;

  v8f acc00 = {0.f, 0.f, 0.f, 0.f, 0.f, 0.f, 0.f, 0.f};
  v8f acc01 = acc00, acc10 = acc00, acc11 = acc00;

  // Per-thread staging slots (constant across K iterations)
  const int ar_ld[4] = {(tid + 0) >> 3, (tid + 256) >> 3,
                        (tid + 512) >> 3, (tid + 768) >> 3};
  const int ac_ld = (tid & 7) << 2;
  const int br_ld[2] = {(tid + 0) >> 4, (tid + 256) >> 4};
  const int bc_ld = (tid & 15) << 2;

  // Prologue: async-stage first K chunk into buffer 0
#pragma unroll
  for (int it = 0; it < 4; ++it) {
    const int r = ar_ld[it];
    async_load_b128(&A[(size_t)(blockM + r) * K + ac_ld], &lsA[0][r][ac_ld]);
  }
#pragma unroll
  for (int it = 0; it < 2; ++it) {
    const int r = br_ld[it];
    async_load_b128(&B[(size_t)r * Nc + blockN + bc_ld], &lsB[0][r][bc_ld]);
  }

  int buf = 0;
  for (int kk0 = 0; kk0 < K; kk0 += BLK_K) {
    wait_async0();       // our slice of the current buffer has landed
    __syncthreads();     // everyone's slice landed; other buffer free to reuse

    // Prefetch next K chunk into the other buffer (overlaps compute below)
    if (kk0 + BLK_K < K) {
      const int nb = buf ^ 1;
      const int kn = kk0 + BLK_K;
#pragma unroll
      for (int it = 0; it < 4; ++it) {
        const int r = ar_ld[it];
        async_load_b128(&A[(size_t)(blockM + r) * K + kn + ac_ld],
                        &lsA[nb][r][ac_ld]);
      }
#pragma unroll
      for (int it = 0; it < 2; ++it) {
        const int r = br_ld[it];
        async_load_b128(&B[(size_t)(kn + r) * Nc + blockN + bc_ld],
                        &lsB[nb][r][bc_ld]);
      }
    }

    const float (*cA)[LDA_S] = lsA[buf];
    const float (*cB)[LDB_S] = lsB[buf];
#pragma unroll
    for (int k = 0; k < BLK_K; k += 4) {
      const int kb = k + 2 * half;
      const int ar = waveRow * 32 + l16;
      const int bc = waveCol * 32 + l16;
      v2f a0, a1, b0, b1;
      a0.x = cA[ar][kb];       a0.y = cA[ar][kb + 1];
      a1.x = cA[ar + 16][kb];  a1.y = cA[ar + 16][kb + 1];
      b0.x = cB[kb][bc];       b0.y = cB[kb + 1][bc];
      b1.x = cB[kb][bc + 16];  b1.y = cB[kb + 1][bc + 16];
      acc00 = __builtin_amdgcn_wmma_f32_16x16x4_f32(false, a0, false, b0,
                                                    (short)0, acc00, false, false);
      acc01 = __builtin_amdgcn_wmma_f32_16x16x4_f32(false, a0, false, b1,
                                                    (short)0, acc01, false, false);
      acc10 = __builtin_amdgcn_wmma_f32_16x16x4_f32(false, a1, false, b0,
                                                    (short)0, acc10, false, false);
      acc11 = __builtin_amdgcn_wmma_f32_16x16x4_f32(false, a1, false, b1,
                                                    (short)0, acc11, false, false);
    }
    buf ^= 1;
  }

  // --- store D (C/D layout: VGPR v -> row v + 8*half, lane -> col) ---
#pragma unroll
  for (int v = 0; v < 8; ++v) {
    const int r0 = blockM + waveRow * 32 + v + 8 * half;
    const int r1 = r0 + 16;
    const int c0 = blockN + waveCol * 32 + l16;
    const int c1 = c0 + 16;
    C[(size_t)r0 * Nc + c0] = acc00[v];
    C[(size_t)r0 * Nc + c1] = acc01[v];
    C[(size_t)r1 * Nc + c0] = acc10[v];
    C[(size_t)r1 * Nc + c1] = acc11[v];
    if (sel != nullptr) {
      if (tsel[r0] == step) {
        sel[(size_t)r0 * Nc + c0] = acc00[v];
        sel[(size_t)r0 * Nc + c1] = acc01[v];
      }
      if (tsel[r1] == step) {
        sel[(size_t)r1 * Nc + c0] = acc10[v];
        sel[(size_t)r1 * Nc + c1] = acc11[v];
      }
    }
  }
}

// ---------------------------------------------------------------------------
// Finalize: x_t = sab[t]*sel + somab[t]*nz ; gather time embeddings.
// ---------------------------------------------------------------------------
__global__ __launch_bounds__(256)
void final_kernel(const float* __restrict__ sel, const float* __restrict__ nz,
                  const float* __restrict__ table, const int* __restrict__ t,
                  const float* __restrict__ sab, const float* __restrict__ somab,
                  float* __restrict__ out) {
  const int idx = blockIdx.x * blockDim.x + threadIdx.x;
  if (idx >= N_NODES * F_DIMS) return;
  const int i = idx >> 9;        // /512
  const int j = idx & 511;
  const int tt = t[i];
  out[idx] = sab[tt] * sel[idx] + somab[tt] * nz[idx];
  out[(size_t)N_NODES * F_DIMS + (size_t)i * H_DIMS + j] =
      table[(size_t)tt * H_DIMS + j];
}

// ---------------------------------------------------------------------------
// Host-side orchestration (graph-capture safe: only kernel launches on stream)
// ---------------------------------------------------------------------------
extern "C" void kernel_launch(void* const* d_in, const int* in_sizes, int n_in,
                              void* d_out, int out_size, void* d_ws, size_t ws_size,
                              hipStream_t stream) {
  const float* x     = (const float*)d_in[0];
  const float* noise = (const float*)d_in[1];
  const float* table = (const float*)d_in[2];
  const int* src     = (const int*)d_in[3];
  const int* dst     = (const int*)d_in[4];
  const int* t       = (const int*)d_in[5];
  float* out = (float*)d_out;
  float* ws = (float*)d_ws;

  const size_t NF = (size_t)N_NODES * F_DIMS;   // 524288
  const size_t NN = (size_t)N_NODES * N_NODES;  // 1048576

  size_t off = 0;
  float* sab   = ws + off; off += T_STEPS;
  float* somab = ws + off; off += T_STEPS;
  float* xn    = ws + off; off += NF;
  float* nz    = ws + off; off += NF;
  float* sel   = ws + off; off += NF;
  float* Amat  = ws + off; off += NN;
  float* Mb0   = ws + off; off += NN;
  float* Mb1   = ws + off; off += NN;
  float* db0   = ws + off; off += NF;
  float* db1   = ws + off; off += NF;

  sched_kernel<<<1, 128, 0, stream>>>(sab, somab);
  ln_kernel<<<N_NODES, 256, 0, stream>>>(x, noise, t, xn, nz, sel);

  zero_kernel<<<(int)(NN / 256), 256, 0, stream>>>(Amat, (int)NN);
  scatter_kernel<<<E_EDGES / 256, 256, 0, stream>>>(src, dst, Amat);
  laplacian_kernel<<<N_NODES, 256, 0, stream>>>(Amat);
  eye_kernel<<<(int)(NN / 256), 256, 0, stream>>>(Mb0);

  dim3 blk(256);
  dim3 gridM(N_NODES / BLK_N, N_NODES / BLK_M);  // (16, 8)
  dim3 gridD(F_DIMS / BLK_N, N_NODES / BLK_M);   // (8, 8)

  float* Mold = Mb0;
  float* Mnew = Mb1;
  const float* dold = xn;
  float* dbufs[2] = {db0, db1};

  for (int step = 1; step < T_STEPS; ++step) {
    // M_t = negL @ M_{t-1}
    wmma_mm_kernel<<<gridM, blk, 0, stream>>>(Mnew, Amat, Mold,
                                              N_NODES, N_NODES, N_NODES,
                                              nullptr, 0, nullptr);
    // d_t = M_t @ d_{t-1}; scatter rows with t[i]==step into sel
    float* dnew = dbufs[(step - 1) & 1];
    wmma_mm_kernel<<<gridD, blk, 0, stream>>>(dnew, Mnew, dold,
                                              N_NODES, F_DIMS, N_NODES,
                                              t, step, sel);
    float* tmp = Mold; Mold = Mnew; Mnew = tmp;
    dold = dnew;
  }

  final_kernel<<<(int)(NF / 256), 256, 0, stream>>>(sel, nz, table, t,
                                                    sab, somab, out);
}